// LinearRouter_74972949119351
// MI455X (gfx1250) — compile-verified
//
#include <hip/hip_runtime.h>

typedef __attribute__((ext_vector_type(2))) float v2f;
typedef __attribute__((ext_vector_type(8))) float v8f;

namespace {
constexpr int kE = 1024;             // hidden dim
constexpr int kM = 64;               // experts
constexpr int kBK = 32;              // K tile
constexpr int kTokensPerBlock = 128; // 8 waves x 16 tokens
constexpr int kThreads = 256;
constexpr int kSeqStride = 36;       // 32 + 4 pad (16B-aligned rows, conflict-free b64 reads)
constexpr int kLogStride = 65;       // 64 + 1 pad (conflict-free per-token rows)
constexpr int kSeqBufFloats = kTokensPerBlock * kSeqStride; // 4608 floats per buffer
constexpr int kNumTiles = kE / kBK;  // 32
}

__global__ __launch_bounds__(kThreads) void router_wmma_kernel(
    const float* __restrict__ seq,
    const float* __restrict__ Wrt,
    const float* __restrict__ bias,
    float* __restrict__ logits_out,
    int* __restrict__ idx_out,
    float* __restrict__ wgt_out)
{
  // phase1: two seq tile buffers (2 x 4608 floats); phase2: logits tile (128*65 = 8320 floats)
  __shared__ float lds[2 * kSeqBufFloats];
  __shared__ float sbias[kM];

  const int tid  = threadIdx.x;
  const int wave = tid >> 5;
  const int lane = tid & 31;
  const int half = lane >> 4;   // K-half selector per WMMA f32 A/B layout
  const int l16  = lane & 15;
  const int koff = half << 1;   // 0 or 2
  const long long tokenBase = (long long)blockIdx.x * kTokensPerBlock;

  if (tid < kM) sbias[tid] = bias[tid];

  v8f acc0 = {}, acc1 = {}, acc2 = {}, acc3 = {};

  const float2* __restrict__ W2 = reinterpret_cast<const float2*>(Wrt);

  // Issue one 128x32 tile's async loads: 4 x b128 per thread, global -> LDS, no VGPR staging.
  auto issueTile = [&](int tile, int buf) {
    const int k0 = tile * kBK;
#pragma unroll
    for (int it = 0; it < 4; ++it) {
      const int fi  = it * kThreads + tid;   // float4 chunk index, 0..1023
      const int row = fi >> 3;               // 8 chunks per 32-float row
      const int col = (fi & 7) << 2;
      const float* g = seq + (tokenBase + row) * kE + k0 + col;
      // low 32 bits of a generic shared pointer == group-segment byte offset
      const unsigned l =
          (unsigned)(unsigned long long)(const void*)&lds[buf * kSeqBufFloats +
                                                          row * kSeqStride + col];
      asm volatile("global_load_async_to_lds_b128 %0, %1, off"
                   :: "v"(l), "v"(g) : "memory");
    }
  };

  issueTile(0, 0);

  for (int t = 0; t < kNumTiles; ++t) {
    const int cur = t & 1;
    if (t + 1 < kNumTiles) {
      issueTile(t + 1, cur ^ 1);                       // prefetch next tile into other buffer
      asm volatile("s_wait_asynccnt 4" ::: "memory");  // in-order: tile t's 4 loads landed
    } else {
      asm volatile("s_wait_asynccnt 0" ::: "memory");  // drain on last tile
    }
    __syncthreads();

    const float* buf = &lds[cur * kSeqBufFloats];
    const int k0 = t * kBK;

#pragma unroll
    for (int kk = 0; kk < kBK; kk += 4) {
      // A fragment: lane holds seq[token = wave*16 + l16][k0+kk+koff .. +1]
      const float2 a2 = *reinterpret_cast<const float2*>(
          &buf[(wave * 16 + l16) * kSeqStride + kk + koff]);
      v2f a; a.x = a2.x; a.y = a2.y;

      const int kglob = k0 + kk + koff;      // even -> valid float2 index

      float2 b0 = W2[((0 * 16 + l16) * kE + kglob) >> 1];
      float2 b1 = W2[((1 * 16 + l16) * kE + kglob) >> 1];
      float2 b2 = W2[((2 * 16 + l16) * kE + kglob) >> 1];
      float2 b3 = W2[((3 * 16 + l16) * kE + kglob) >> 1];

      v2f bv0; bv0.x = b0.x; bv0.y = b0.y;
      v2f bv1; bv1.x = b1.x; bv1.y = b1.y;
      v2f bv2; bv2.x = b2.x; bv2.y = b2.y;
      v2f bv3; bv3.x = b3.x; bv3.y = b3.y;

      acc0 = __builtin_amdgcn_wmma_f32_16x16x4_f32(false, a, false, bv0, (short)0, acc0, false, false);
      acc1 = __builtin_amdgcn_wmma_f32_16x16x4_f32(false, a, false, bv1, (short)0, acc1, false, false);
      acc2 = __builtin_amdgcn_wmma_f32_16x16x4_f32(false, a, false, bv2, (short)0, acc2, false, false);
      acc3 = __builtin_amdgcn_wmma_f32_16x16x4_f32(false, a, false, bv3, (short)0, acc3, false, false);
    }
    __syncthreads();
  }

  // ---- scatter accumulators: global logits (non-temporal) + LDS logits tile ----
  // C/D layout: VGPR j -> row M = j (lanes 0-15) or j+8 (lanes 16-31), col N = l16
  {
    const int rbase = wave * 16 + half * 8;
#pragma unroll
    for (int j = 0; j < 8; ++j) {
      const int r = rbase + j;
      const long long tok = tokenBase + r;
      float* gout = logits_out + tok * kM;
      __builtin_nontemporal_store(acc0[j], &gout[ 0 + l16]);
      __builtin_nontemporal_store(acc1[j], &gout[16 + l16]);
      __builtin_nontemporal_store(acc2[j], &gout[32 + l16]);
      __builtin_nontemporal_store(acc3[j], &gout[48 + l16]);
      float* lrow = &lds[r * kLogStride];
      lrow[ 0 + l16] = acc0[j];
      lrow[16 + l16] = acc1[j];
      lrow[32 + l16] = acc2[j];
      lrow[48 + l16] = acc3[j];
    }
  }
  __syncthreads();

  // ---- fused softmax + top-2 + normalize (one thread per token) ----
  if (tid < kTokensPerBlock) {
    const float* row = &lds[tid * kLogStride];

    float mx = -3.402823466e+38f;
    for (int i = 0; i < kM; ++i) mx = fmaxf(mx, row[i]);

    float s = 0.f;
    for (int i = 0; i < kM; ++i) s += __expf(row[i] - mx);
    const float inv = 1.0f / s;

    float bestA = -3.402823466e+38f, secA = -3.402823466e+38f;
    float bestP = 0.f, secP = 0.f;
    int   bestI = 0,   secI = 0;
    for (int i = 0; i < kM; ++i) {
      const float p   = __expf(row[i] - mx) * inv; // softmax prob
      const float adj = p + sbias[i];              // selection score
      if (adj > bestA) {
        secA = bestA; secP = bestP; secI = bestI;
        bestA = adj;  bestP = p;    bestI = i;
      } else if (adj > secA) {
        secA = adj; secP = p; secI = i;
      }
    }
    const float denom = bestP + secP + 1e-6f;
    const long long tok = tokenBase + tid;
    idx_out[tok * 2 + 0] = bestI;
    idx_out[tok * 2 + 1] = secI;
    wgt_out[tok * 2 + 0] = bestP / denom;
    wgt_out[tok * 2 + 1] = secP / denom;
  }
}

extern "C" void kernel_launch(void* const* d_in, const int* in_sizes, int n_in,
                              void* d_out, int out_size, void* d_ws, size_t ws_size,
                              hipStream_t stream) {
  const float* seq  = (const float*)d_in[0];
  const float* W    = (const float*)d_in[1];
  const float* bias = (const float*)d_in[2];

  const long long nTokens = (long long)in_sizes[0] / kE; // B*N = 32768

  float* logits = (float*)d_out;
  int*   idxs   = (int*)((float*)d_out + nTokens * kM);
  float* wgts   = (float*)d_out + nTokens * kM + nTokens * 2;

  dim3 grid((unsigned)(nTokens / kTokensPerBlock));
  router_wmma_kernel<<<grid, kThreads, 0, stream>>>(seq, W, bias, logits, idxs, wgts);
}